// GraphConv_Edge_Residual_32031866093817
// MI455X (gfx1250) — compile-verified
//
#include <hip/hip_runtime.h>
#include <math.h>

typedef float v2f __attribute__((ext_vector_type(2)));
typedef float v8f __attribute__((ext_vector_type(8)));

__device__ __forceinline__ v8f wmma_f32(v2f a, v2f b, v8f c) {
  // V_WMMA_F32_16X16X4_F32 : D = A(16x4) * B(4x16) + C(16x16)
  return __builtin_amdgcn_wmma_f32_16x16x4_f32(false, a, false, b, (short)0, c,
                                               false, false);
}

// ---------------------------------------------------------------------------
// Utility kernels
// ---------------------------------------------------------------------------
__global__ void zero_kernel(float* __restrict__ p, int n) {
  int i = blockIdx.x * blockDim.x + threadIdx.x;
  if (i < n) p[i] = 0.0f;
}

__global__ void degree_kernel(const int* __restrict__ src,
                              const int* __restrict__ dst,
                              float* __restrict__ outdeg,
                              float* __restrict__ indeg, int E) {
  int e = blockIdx.x * blockDim.x + threadIdx.x;
  if (e < E) {
    unsafeAtomicAdd(&outdeg[src[e]], 1.0f);
    unsafeAtomicAdd(&indeg[dst[e]], 1.0f);
  }
}

__global__ void scale_kernel(const float* __restrict__ outdeg,
                             const float* __restrict__ indeg,
                             float* __restrict__ outscale,
                             float* __restrict__ inscale, int N) {
  int i = blockIdx.x * blockDim.x + threadIdx.x;
  if (i < N) {
    outscale[i] = rsqrtf(fmaxf(outdeg[i], 1.0f));
    inscale[i]  = rsqrtf(fmaxf(indeg[i], 1.0f));
  }
}

// ---------------------------------------------------------------------------
// Generic GEMM: Y = X @ W + b.   D = 256 fixed.
// Block = 128 threads (4 waves). Block tile: 64 rows x 256 cols.
// Wave tile: 64 rows x 64 cols = 4 M-subtiles x 4 N-tiles (16 v8f accums).
// B fragments are shared across the 4 M-subtiles -> 16 WMMA per 12 loads.
// ---------------------------------------------------------------------------
__global__ __launch_bounds__(128) void gemm_bias(
    const float* __restrict__ X, const float* __restrict__ W,
    const float* __restrict__ bvec, float* __restrict__ Y, int nrows) {
  const int lane = threadIdx.x & 31;
  const int wave = threadIdx.x >> 5;
  const int row0 = blockIdx.x * 64;
  const int col0 = wave * 64;
  const int half = lane >> 4;  // 0: K=0,1 / M=r ; 1: K=2,3 / M=r+8
  const int l16  = lane & 15;
  const int koff = half * 2;

  int arow[4];
#pragma unroll
  for (int s = 0; s < 4; ++s)
    arow[s] = min(row0 + s * 16 + l16, nrows - 1);

  v8f acc[4][4] = {};  // [m-subtile][n-tile]

  for (int k0 = 0; k0 < 256; k0 += 4) {
    v2f a[4];
#pragma unroll
    for (int s = 0; s < 4; ++s)
      a[s] = *(const v2f*)(X + (size_t)arow[s] * 256 + k0 + koff);
#pragma unroll
    for (int t = 0; t < 4; ++t) {
      const int n = col0 + t * 16 + l16;
      v2f b;
      b.x = W[(size_t)(k0 + koff) * 256 + n];
      b.y = W[(size_t)(k0 + koff + 1) * 256 + n];
#pragma unroll
      for (int s = 0; s < 4; ++s) acc[s][t] = wmma_f32(a[s], b, acc[s][t]);
    }
  }

#pragma unroll
  for (int s = 0; s < 4; ++s) {
#pragma unroll
    for (int t = 0; t < 4; ++t) {
      const int n = col0 + t * 16 + l16;
#pragma unroll
      for (int r = 0; r < 8; ++r) {
        const int m = row0 + s * 16 + r + half * 8;
        if (m < nrows) Y[(size_t)m * 256 + n] = acc[s][t][r] + bvec[n];
      }
    }
  }
}

// ---------------------------------------------------------------------------
// Fused edge pipeline (64 edges x 256 cols per block):
//   g   = edge_feats @ Wedge                            (WMMA)
//   m   = g + bedge[n] + Esrc[src[e]] + Edst[dst[e]]    -> m_out
//   msg = X[src[e]] * outscale[src[e]] * sigmoid(m)
//   agg[dst[e]] += msg                                  (L2 atomic fadd)
// ---------------------------------------------------------------------------
__global__ __launch_bounds__(128) void edge_gate_kernel(
    const float* __restrict__ Ef, const float* __restrict__ We,
    const float* __restrict__ be, const float* __restrict__ Esrc,
    const float* __restrict__ Edst, const float* __restrict__ X,
    const float* __restrict__ outscale, const int* __restrict__ src,
    const int* __restrict__ dst, float* __restrict__ m_out,
    float* __restrict__ agg, int E) {
  const int lane = threadIdx.x & 31;
  const int wave = threadIdx.x >> 5;
  const int e0   = blockIdx.x * 64;
  const int col0 = wave * 64;
  const int half = lane >> 4;
  const int l16  = lane & 15;
  const int koff = half * 2;

  int arow[4];
#pragma unroll
  for (int s = 0; s < 4; ++s)
    arow[s] = min(e0 + s * 16 + l16, E - 1);

  v8f acc[4][4] = {};

  for (int k0 = 0; k0 < 256; k0 += 4) {
    v2f a[4];
#pragma unroll
    for (int s = 0; s < 4; ++s)
      a[s] = *(const v2f*)(Ef + (size_t)arow[s] * 256 + k0 + koff);
#pragma unroll
    for (int t = 0; t < 4; ++t) {
      const int n = col0 + t * 16 + l16;
      v2f b;
      b.x = We[(size_t)(k0 + koff) * 256 + n];
      b.y = We[(size_t)(k0 + koff + 1) * 256 + n];
#pragma unroll
      for (int s = 0; s < 4; ++s) acc[s][t] = wmma_f32(a[s], b, acc[s][t]);
    }
  }

  // Epilogue: one M-subtile (16 edges) at a time to bound register use.
#pragma unroll
  for (int s = 0; s < 4; ++s) {
    const int eb = e0 + s * 16;
    int sidx[8], didx[8];
    float osc[8];
#pragma unroll
    for (int r = 0; r < 8; ++r) {
      const int e = min(eb + r + half * 8, E - 1);
      sidx[r] = src[e];
      didx[r] = dst[e];
      osc[r]  = outscale[sidx[r]];
    }
#pragma unroll
    for (int t = 0; t < 4; ++t) {
      const int n = col0 + t * 16 + l16;
      const float bn = be[n];
#pragma unroll
      for (int r = 0; r < 8; ++r) {
        const int e = eb + r + half * 8;
        if (e < E) {
          const int sn = sidx[r];
          const int dn = didx[r];
          const float mv = acc[s][t][r] + bn + Esrc[(size_t)sn * 256 + n] +
                           Edst[(size_t)dn * 256 + n];
          m_out[(size_t)e * 256 + n] = mv;
          const float sig = 1.0f / (1.0f + __expf(-mv));
          const float msg = X[(size_t)sn * 256 + n] * osc[r] * sig;
          unsafeAtomicAdd(&agg[(size_t)dn * 256 + n], msg);
        }
      }
    }
  }
}

// ---------------------------------------------------------------------------
// rst = (agg @ W) * inscale + bias + X   (residual projection)
// ---------------------------------------------------------------------------
__global__ __launch_bounds__(128) void out_gemm(
    const float* __restrict__ Agg, const float* __restrict__ W,
    const float* __restrict__ bias, const float* __restrict__ X,
    const float* __restrict__ inscale, float* __restrict__ out, int nrows) {
  const int lane = threadIdx.x & 31;
  const int wave = threadIdx.x >> 5;
  const int row0 = blockIdx.x * 64;
  const int col0 = wave * 64;
  const int half = lane >> 4;
  const int l16  = lane & 15;
  const int koff = half * 2;

  int arow[4];
#pragma unroll
  for (int s = 0; s < 4; ++s)
    arow[s] = min(row0 + s * 16 + l16, nrows - 1);

  v8f acc[4][4] = {};

  for (int k0 = 0; k0 < 256; k0 += 4) {
    v2f a[4];
#pragma unroll
    for (int s = 0; s < 4; ++s)
      a[s] = *(const v2f*)(Agg + (size_t)arow[s] * 256 + k0 + koff);
#pragma unroll
    for (int t = 0; t < 4; ++t) {
      const int n = col0 + t * 16 + l16;
      v2f b;
      b.x = W[(size_t)(k0 + koff) * 256 + n];
      b.y = W[(size_t)(k0 + koff + 1) * 256 + n];
#pragma unroll
      for (int s = 0; s < 4; ++s) acc[s][t] = wmma_f32(a[s], b, acc[s][t]);
    }
  }

#pragma unroll
  for (int s = 0; s < 4; ++s) {
#pragma unroll
    for (int t = 0; t < 4; ++t) {
      const int n = col0 + t * 16 + l16;
      const float bn = bias[n];
#pragma unroll
      for (int r = 0; r < 8; ++r) {
        const int m = row0 + s * 16 + r + half * 8;
        if (m < nrows) {
          out[(size_t)m * 256 + n] =
              acc[s][t][r] * inscale[m] + bn + X[(size_t)m * 256 + n];
        }
      }
    }
  }
}

// ---------------------------------------------------------------------------
// Launcher
// ---------------------------------------------------------------------------
extern "C" void kernel_launch(void* const* d_in, const int* in_sizes, int n_in,
                              void* d_out, int out_size, void* d_ws,
                              size_t ws_size, hipStream_t stream) {
  const float* node_feats = (const float*)d_in[0];
  const float* edge_feats = (const float*)d_in[1];
  const int*   src        = (const int*)d_in[2];
  const int*   dst        = (const int*)d_in[3];
  const float* weight     = (const float*)d_in[4];
  const float* bias       = (const float*)d_in[5];
  const float* w_src      = (const float*)d_in[6];
  const float* b_src      = (const float*)d_in[7];
  const float* w_dst      = (const float*)d_in[8];
  const float* b_dst      = (const float*)d_in[9];
  const float* w_edge     = (const float*)d_in[10];
  const float* b_edge     = (const float*)d_in[11];

  const int D = in_sizes[5];      // 256
  const int N = in_sizes[0] / D;  // 10000
  const int E = in_sizes[2];      // 320000

  float* out   = (float*)d_out;
  float* rst   = out;                  // [N, D]
  float* m_out = out + (size_t)N * D;  // [E, D]

  // Workspace layout (floats):
  float* ws       = (float*)d_ws;
  float* Esrc     = ws;                    // N*D
  float* Edst     = Esrc + (size_t)N * D;  // N*D
  float* agg      = Edst + (size_t)N * D;  // N*D
  float* outdeg   = agg + (size_t)N * D;   // N
  float* indeg    = outdeg + N;            // N
  float* outscale = indeg + N;             // N
  float* inscale  = outscale + N;          // N

  // 1) zero agg + degree counters (contiguous region)
  {
    int nz = N * D + 2 * N;
    zero_kernel<<<(nz + 255) / 256, 256, 0, stream>>>(agg, nz);
  }

  // 2) degrees
  degree_kernel<<<(E + 255) / 256, 256, 0, stream>>>(src, dst, outdeg, indeg,
                                                     E);

  // 3) normalization scales
  scale_kernel<<<(N + 255) / 256, 256, 0, stream>>>(outdeg, indeg, outscale,
                                                    inscale, N);

  // 4) node gate projections (e_src, e_dst)
  {
    int mtiles = (N + 63) / 64;
    gemm_bias<<<mtiles, 128, 0, stream>>>(node_feats, w_src, b_src, Esrc, N);
    gemm_bias<<<mtiles, 128, 0, stream>>>(node_feats, w_dst, b_dst, Edst, N);
  }

  // 5) fused edge gate GEMM + sigmoid + scatter aggregation
  {
    int etiles = (E + 63) / 64;
    edge_gate_kernel<<<etiles, 128, 0, stream>>>(edge_feats, w_edge, b_edge,
                                                 Esrc, Edst, node_feats,
                                                 outscale, src, dst, m_out, agg,
                                                 E);
  }

  // 6) output projection + degree norm + bias + residual
  {
    int mtiles = (N + 63) / 64;
    out_gemm<<<mtiles, 128, 0, stream>>>(agg, weight, bias, node_feats, inscale,
                                         rst, N);
  }
}